// TransitionDownBlock_2173253452356
// MI455X (gfx1250) — compile-verified
//
#include <hip/hip_runtime.h>
#include <hip/hip_bf16.h>

typedef __attribute__((ext_vector_type(2))) float v2f;
typedef __attribute__((ext_vector_type(8))) float v8f;

#define INF_F 3.4028235e38f

// ---------------------------------------------------------------------------
// GEMM via V_WMMA_F32_16X16X4_F32:  Y[b,o,n] = sum_c W[o,c] * X[b,c,n]
//
// One wave32 per (batch, 16-column) strip covering ALL Cout rows:
//   - X fragment (B matrix, 4x16) loaded once per K-step, reused by MT=8 WMMAs
//   - W staged in LDS (padded stride Cin+2 -> conflict-free ds_load_2addr_b32)
//   - A-fragments loaded as a BATCH into distinct registers before the WMMA
//     burst, so ds_loads clause together with graduated dscnt waits instead of
//     one full s_wait_dscnt 0 stall per WMMA
//   - 8 independent v8f accumulators (no WMMA->WMMA RAW hazard on D)
//   - steady-state loop is branch-free (last K-step peeled for the prefetch
//     guard); running pointers avoid per-iteration 64-bit address muls
// A 16x4 layout: lanes 0-15 hold M=0..15 {K0,K1}; lanes 16-31 hold {K2,K3}
// C/D 16x16 layout: VGPR r: lanes 0-15 -> M=r, lanes 16-31 -> M=r+8
// ---------------------------------------------------------------------------
template <int MT>
__global__ __launch_bounds__(256)
void gemm_wmma_f32(const float* __restrict__ W, const float* __restrict__ X,
                   float* __restrict__ Y, int Cout, int Cin, int N) {
    extern __shared__ float sW[];              // Cout * (Cin + 2), bank-padded
    const int tid  = threadIdx.x;
    const int lane = tid & 31;
    const int wid  = tid >> 5;
    const int CS   = Cin + 2;

    // cooperative stage of the (tiny, fully reused) weight matrix into LDS
    for (int i = tid; i < Cout * Cin; i += 256) {
        const int row = i / Cin;
        const int c   = i - row * Cin;
        sW[row * CS + c] = W[i];
    }
    __syncthreads();

    const int tilesN = N >> 4;
    const int strip  = blockIdx.x * 8 + wid;   // (b, nt) column strip
    const int b      = strip / tilesN;
    const int nt     = strip - b * tilesN;

    const int col   = nt * 16 + (lane & 15);
    const int khalf = (lane >> 4) * 2;         // 0 or 2
    const long long sN = (long long)N;

    v8f acc[MT];
#pragma unroll
    for (int mt = 0; mt < MT; ++mt) acc[mt] = (v8f){};

    // running pointers: one add per operand per K-step, no in-loop muls
    const float* xptr  = X + (long long)b * Cin * N + col + (long long)khalf * sN;
    const float* wbase = sW + (lane & 15) * CS + khalf;

    auto kstep = [&](const float* xp, const float* wk) {
        v2f bf;
        bf.x = xp[0];
        bf.y = xp[sN];
        // batch-load all A fragments first (distinct regs -> clause + graduated
        // dscnt waits), then fire the 8-WMMA burst
        v2f afrag[MT];
#pragma unroll
        for (int mt = 0; mt < MT; ++mt) {
            afrag[mt].x = wk[mt * 16 * CS];
            afrag[mt].y = wk[mt * 16 * CS + 1];
        }
#pragma unroll
        for (int mt = 0; mt < MT; ++mt) {
            acc[mt] = __builtin_amdgcn_wmma_f32_16x16x4_f32(
                /*neg_a=*/false, afrag[mt], /*neg_b=*/false, bf,
                /*c_mod=*/(short)0, acc[mt], /*reuse_a=*/false, /*reuse_b=*/false);
        }
    };

    int wofs = 0;
    for (int k0 = 0; k0 < Cin - 4; k0 += 4) {
        __builtin_prefetch(xptr + 4 * sN, 0, 1);   // gfx1250 global_prefetch_b8
        kstep(xptr, wbase + wofs);
        xptr += 4 * sN;
        wofs += 4;
    }
    kstep(xptr, wbase + wofs);                     // peeled final K-step

    const int msub = (lane >> 4) * 8;
    float* Yb = Y + (long long)b * Cout * N + col;
#pragma unroll
    for (int mt = 0; mt < MT; ++mt) {
        float* Yo = Yb + (long long)(mt * 16 + msub) * N;
#pragma unroll
        for (int rr = 0; rr < 8; ++rr) Yo[(long long)rr * N] = acc[mt][rr];
    }
}

// ---------------------------------------------------------------------------
// BatchNorm statistics (training mode: mean/var over B,N per channel)
// ---------------------------------------------------------------------------
__global__ __launch_bounds__(256)
void bn_stats(const float* __restrict__ Y, float* __restrict__ mean,
              float* __restrict__ invstd, int Cout, int N, int B) {
    const int o = blockIdx.x;
    const int tid = threadIdx.x;
    float s = 0.f, s2 = 0.f;
    for (int b = 0; b < B; ++b) {
        const float* p = Y + ((long long)b * Cout + o) * N;
        for (int n = tid; n < N; n += 256) { float v = p[n]; s += v; s2 += v * v; }
    }
    __shared__ float sh[256], sh2[256];
    sh[tid] = s; sh2[tid] = s2;
    __syncthreads();
    for (int off = 128; off > 0; off >>= 1) {
        if (tid < off) { sh[tid] += sh[tid + off]; sh2[tid] += sh2[tid + off]; }
        __syncthreads();
    }
    if (tid == 0) {
        const float cnt = (float)B * (float)N;
        const float mu  = sh[0] / cnt;
        const float var = sh2[0] / cnt - mu * mu;
        mean[o]   = mu;
        invstd[o] = rsqrtf(var + 1e-5f);
    }
}

// ---------------------------------------------------------------------------
// In-place normalize + affine + ReLU
// ---------------------------------------------------------------------------
__global__ __launch_bounds__(256)
void bn_relu(float* __restrict__ Y, const float* __restrict__ mean,
             const float* __restrict__ invstd, const float* __restrict__ gamma,
             const float* __restrict__ beta, int Cout, int N, long long total) {
    long long i = (long long)blockIdx.x * 256 + threadIdx.x;
    if (i >= total) return;
    int o = (int)((i / N) % Cout);
    float v = (Y[i] - mean[o]) * invstd[o] * gamma[o] + beta[o];
    Y[i] = fmaxf(v, 0.f);
}

// ---------------------------------------------------------------------------
// Furthest point sampling: one block per batch, sequential argmax-min chain.
// 1024 threads * 8 points each, all state in registers; 2-level wave32 reduce.
// ---------------------------------------------------------------------------
__global__ __launch_bounds__(1024)
void fps_kernel(const float* __restrict__ pts, int* __restrict__ idx,
                float* __restrict__ cen, int N, int M) {
    const int b = blockIdx.x;
    const int tid = threadIdx.x;
    const int lane = tid & 31;
    const int wid = tid >> 5;
    const float* px = pts + (long long)(b * 3 + 0) * N;
    const float* py = pts + (long long)(b * 3 + 1) * N;
    const float* pz = pts + (long long)(b * 3 + 2) * N;

    float lx[8], ly[8], lz[8], dmin[8];
#pragma unroll
    for (int j = 0; j < 8; ++j) {
        const int n = tid + j * 1024;
        lx[j] = px[n]; ly[j] = py[n]; lz[j] = pz[n];
        dmin[j] = INF_F;
    }

    __shared__ float qs[3];
    __shared__ float wd[32];
    __shared__ int   wi[32];
    if (tid == 0) {
        idx[(long long)b * M] = 0;
        qs[0] = px[0]; qs[1] = py[0]; qs[2] = pz[0];
    }
    __syncthreads();

    for (int it = 1; it < M; ++it) {
        const float qx = qs[0], qy = qs[1], qz = qs[2];
        float best = -1.f; int bestI = 0;
#pragma unroll
        for (int j = 0; j < 8; ++j) {
            const float dx = lx[j] - qx, dy = ly[j] - qy, dz = lz[j] - qz;
            float d = dx * dx + dy * dy + dz * dz;
            d = fminf(d, dmin[j]);
            dmin[j] = d;
            if (d > best) { best = d; bestI = tid + j * 1024; }
        }
        // wave32 argmax butterfly (tie -> smaller index, matching argmax-first)
        for (int off = 16; off > 0; off >>= 1) {
            const float od = __shfl_xor(best, off, 32);
            const int   oi = __shfl_xor(bestI, off, 32);
            if (od > best || (od == best && oi < bestI)) { best = od; bestI = oi; }
        }
        if (lane == 0) { wd[wid] = best; wi[wid] = bestI; }
        __syncthreads();
        if (tid < 32) {
            float d = wd[tid]; int i2 = wi[tid];
            for (int off = 16; off > 0; off >>= 1) {
                const float od = __shfl_xor(d, off, 32);
                const int   oi = __shfl_xor(i2, off, 32);
                if (od > d || (od == d && oi < i2)) { d = od; i2 = oi; }
            }
            if (tid == 0) {
                idx[(long long)b * M + it] = i2;
                qs[0] = px[i2]; qs[1] = py[i2]; qs[2] = pz[i2];
            }
        }
        __syncthreads();
    }

    // gather centroids [B,3,M] into output
    for (int i = tid; i < M; i += 1024) {
        const int ii = idx[(long long)b * M + i];
        cen[(long long)(b * 3 + 0) * M + i] = px[ii];
        cen[(long long)(b * 3 + 1) * M + i] = py[ii];
        cen[(long long)(b * 3 + 2) * M + i] = pz[ii];
    }
}

// ---------------------------------------------------------------------------
// KNN (k=16) + neighbor mean pooling. One wave32 per centroid.
// Register-resident top-16 via replace-max; 16-round cross-lane merge.
// ---------------------------------------------------------------------------
__global__ __launch_bounds__(128)
void knn_pool(const float* __restrict__ pts, const float* __restrict__ cen,
              const float* __restrict__ F, float* __restrict__ out,
              int N, int M, int C) {
    const int lane = threadIdx.x & 31;
    const int wid  = threadIdx.x >> 5;
    const int wg = blockIdx.x * 4 + wid;
    const int b = wg / M;
    const int m = wg - b * M;

    const float cx = cen[(long long)(b * 3 + 0) * M + m];
    const float cy = cen[(long long)(b * 3 + 1) * M + m];
    const float cz = cen[(long long)(b * 3 + 2) * M + m];
    const float* px = pts + (long long)(b * 3 + 0) * N;
    const float* py = pts + (long long)(b * 3 + 1) * N;
    const float* pz = pts + (long long)(b * 3 + 2) * N;

    float bd[16]; int bi[16];
#pragma unroll
    for (int j = 0; j < 16; ++j) { bd[j] = INF_F; bi[j] = 0x7fffffff; }
    float wmax = INF_F; int wpos = 0;

    for (int n = lane; n < N; n += 32) {
        const float dx = px[n] - cx, dy = py[n] - cy, dz = pz[n] - cz;
        const float d = dx * dx + dy * dy + dz * dz;
        if (d < wmax) {
#pragma unroll
            for (int j = 0; j < 16; ++j) if (j == wpos) { bd[j] = d; bi[j] = n; }
            wmax = bd[0]; wpos = 0;
#pragma unroll
            for (int j = 1; j < 16; ++j) if (bd[j] > wmax) { wmax = bd[j]; wpos = j; }
        }
    }

    __shared__ int nbr[4][16];
    for (int r = 0; r < 16; ++r) {
        float lm = bd[0]; int lidx = bi[0]; int lp = 0;
#pragma unroll
        for (int j = 1; j < 16; ++j)
            if (bd[j] < lm || (bd[j] == lm && bi[j] < lidx)) { lm = bd[j]; lidx = bi[j]; lp = j; }
        float md = lm; int midx = lidx; int mkey = (lane << 4) | lp;
        for (int off = 16; off > 0; off >>= 1) {
            const float od = __shfl_xor(md, off, 32);
            const int   oi = __shfl_xor(midx, off, 32);
            const int   ok = __shfl_xor(mkey, off, 32);
            if (od < md || (od == md && oi < midx)) { md = od; midx = oi; mkey = ok; }
        }
        if ((mkey >> 4) == lane) {
            const int s = mkey & 15;
#pragma unroll
            for (int j = 0; j < 16; ++j) if (j == s) bd[j] = INF_F;
        }
        if (lane == 0) nbr[wid][r] = midx;
    }
    __syncthreads();

    const float* Fb = F + (long long)b * C * N;
    float* ob = out + (long long)b * C * M;
    for (int c = lane; c < C; c += 32) {
        const float* Fc = Fb + (long long)c * N;
        float acc = 0.f;
#pragma unroll
        for (int k = 0; k < 16; ++k) acc += Fc[nbr[wid][k]];
        ob[(long long)c * M + m] = acc * (1.0f / 16.0f);
    }
}

// ---------------------------------------------------------------------------
extern "C" void kernel_launch(void* const* d_in, const int* in_sizes, int n_in,
                              void* d_out, int out_size, void* d_ws, size_t ws_size,
                              hipStream_t stream) {
    (void)n_in; (void)out_size; (void)ws_size;
    const float* feats  = (const float*)d_in[0];
    const float* points = (const float*)d_in[1];
    const float* W1     = (const float*)d_in[2];
    const float* gamma1 = (const float*)d_in[3];
    const float* beta1  = (const float*)d_in[4];
    const float* W2     = (const float*)d_in[5];
    const float* gamma2 = (const float*)d_in[6];
    const float* beta2  = (const float*)d_in[7];

    const int B    = 4;
    const int Cout = in_sizes[3];           // 128
    const int Cin  = in_sizes[2] / Cout;    // 64
    const int N    = in_sizes[1] / (3 * B); // 8192
    const int M    = (N + 3) / 4;           // downsample_ratio = 4 -> 2048

    float* Y1     = (float*)d_ws;
    float* Y2     = Y1 + (size_t)B * Cout * N;
    float* meanv  = Y2 + (size_t)B * Cout * N;
    float* invstd = meanv + Cout;
    int*   idx    = (int*)(invstd + Cout);

    float* outFeat = (float*)d_out;                   // [B, Cout, M]
    float* cenOut  = outFeat + (size_t)B * Cout * M;  // [B, 3, M]

    const int strips = B * (N / 16);                  // 2048 column strips
    const long long tot = (long long)B * Cout * N;

    // layer 1: W1 @ feats -> BN -> ReLU  (in place in Y1)
    gemm_wmma_f32<8><<<strips / 8, 256, (size_t)Cout * (Cin + 2) * 4, stream>>>(
        W1, feats, Y1, Cout, Cin, N);
    bn_stats<<<Cout, 256, 0, stream>>>(Y1, meanv, invstd, Cout, N, B);
    bn_relu<<<(int)((tot + 255) / 256), 256, 0, stream>>>(Y1, meanv, invstd, gamma1, beta1, Cout, N, tot);

    // layer 2: W2 @ F1 -> BN -> ReLU  (in place in Y2)
    gemm_wmma_f32<8><<<strips / 8, 256, (size_t)Cout * (Cout + 2) * 4, stream>>>(
        W2, Y1, Y2, Cout, Cout, N);
    bn_stats<<<Cout, 256, 0, stream>>>(Y2, meanv, invstd, Cout, N, B);
    bn_relu<<<(int)((tot + 255) / 256), 256, 0, stream>>>(Y2, meanv, invstd, gamma2, beta2, Cout, N, tot);

    // furthest point sampling + centroid gather (centroids straight into d_out)
    fps_kernel<<<B, 1024, 0, stream>>>(points, idx, cenOut, N, M);

    // kNN (k=16) + mean pooling of transformed features
    knn_pool<<<(B * M) / 4, 128, 0, stream>>>(points, cenOut, Y2, outFeat, N, M, Cout);
}